// CFPRF_PRN_8340826489608
// MI455X (gfx1250) — compile-verified
//
#include <hip/hip_runtime.h>
#include <math.h>

typedef __attribute__((ext_vector_type(2))) float v2f;
typedef __attribute__((ext_vector_type(8))) float v8f;

#define DEVFN __device__ __forceinline__

namespace {
constexpr int T_ = 2048;
constexpr int F_ = 576;
constexpr int L_ = 192;

// ---- block1 LDS (floats) ----
constexpr int B1_SPRE  = 32 * 288;        // prepool row; first 32*192 reused as conv2-A staging
constexpr int B1_SELU  = 2 * 32 * 292;    // rolling 2 selu rows, channel-pair interleaved
constexpr int B1_SX    = 3 * 580;         // rolling 3 input rows, padded +-2
constexpr int B1_SMALL = 32 * 4 + 96 + 192;
constexpr int B1_LDS   = B1_SPRE + B1_SELU + B1_SX + B1_SMALL;
constexpr int PW1 = 292 * 2;              // pair-row stride (floats) in block1 selu

// ---- block2 LDS (floats) ----
constexpr int B2_SA1   = 64 * 192;        // conv1 A (BN-scale folded)
constexpr int B2_SAD   = 64 * 96;         // downsample A
constexpr int B2_SELU  = 2 * 64 * 196;    // rolling 2 selu rows, channel-pair interleaved
constexpr int B2_SY    = 2 * 32 * 196;    // rolling 2 y1 rows, channel-pair interleaved
constexpr int B2_SPRE  = 64 * 192;        // prepool row
constexpr int B2_SMALL = 64 * 3 + 64;     // bnB, b2, bd, featAcc
constexpr int B2_LDS   = B2_SA1 + B2_SAD + B2_SELU + B2_SY + B2_SPRE + B2_SMALL;
constexpr int PW2 = 196 * 2;              // pair-row stride (floats) in block2 buffers

constexpr size_t Y1_FLOATS = (size_t)32 * 32 * 192 * 192;
} // namespace

DEVFN v8f wmma4(v2f a, v2f b, v8f c) {
  // D = A(16x4) * B(4x16) + C, f32 in/out
  return __builtin_amdgcn_wmma_f32_16x16x4_f32(false, a, false, b, (short)0, c,
                                               false, false);
}

DEVFN v2f ld2(const float* p) { return *reinterpret_cast<const v2f*>(p); }

DEVFN float selu_f(float x) {
  const float al = 1.6732632423543772f, sc = 1.0507009873554805f;
  return sc * (x > 0.f ? x : al * (__expf(x) - 1.f));
}

// ============================ Block 1 ============================
// 1->32 ch, H=192, W=576; conv1(2x3,pad1,1)+BN+SELU -> conv2(2x3,pad0,1)
// + downsample(1x3) -> maxpool W/3 -> masked store y1 (32,32,192,192).
__global__ __launch_bounds__(256) void k_block1(
    const float* __restrict__ embs, const int* __restrict__ bidx,
    const int* __restrict__ tidx, const int* __restrict__ rlen,
    const float* __restrict__ w1, const float* __restrict__ b1,
    const float* __restrict__ bg, const float* __restrict__ bb,
    const float* __restrict__ bm, const float* __restrict__ bv,
    const float* __restrict__ w2, const float* __restrict__ b2,
    const float* __restrict__ wd, const float* __restrict__ bd,
    float* __restrict__ y1) {
  extern __shared__ float sm1[];
  float* sP  = sm1;                 // [32][288]   (first reused as A staging [32][192])
  float* sS  = sP + B1_SPRE;        // 2 slots, pair-interleaved: (ci>>1)*PW1 + 2*s + (ci&1)
  float* sX  = sS + B1_SELU;        // [3][580]
  float* bnA = sX + B1_SX;
  float* bnB = bnA + 32;
  float* b2f = bnB + 32;
  float* bdf = b2f + 32;
  float* wdf = bdf + 32;            // [32][3]
  float* w1f = wdf + 96;            // [32][6]

  const int tid  = threadIdx.x;
  const int lane = tid & 31;
  const int wave = tid >> 5;
  const int bid   = blockIdx.x;
  const int p     = bid / 12;
  const int rem   = bid % 12;
  const int h0    = (rem >> 1) * 32;
  const int wbase = (rem & 1) * 288;
  const int len   = rlen[p];

  for (int i = tid; i < 32; i += 256) {
    float sc = bg[i] / sqrtf(bv[i] + 1e-5f);
    bnA[i] = sc;
    bnB[i] = (b1[i] - bm[i]) * sc + bb[i];
    b2f[i] = b2[i];
    bdf[i] = bd[i];
  }
  for (int i = tid; i < 96; i += 256) wdf[i] = wd[i];
  for (int i = tid; i < 192; i += 256) w1f[i] = w1[i];
  // conv2 A staging: k = kw*64 + kh*32 + ci
  for (int i = tid; i < 32 * 192; i += 256) {
    int co = i / 192, k = i % 192;
    int ci = k & 31, kh = (k >> 5) & 1, kw = k >> 6;
    sP[i] = w2[((co * 32 + ci) * 2 + kh) * 3 + kw];
  }
  __syncthreads();

  const int mtile = wave & 1;
  const int am    = mtile * 16 + (lane & 15);
  const int khalf = lane >> 4;
  v2f aF[48];
#pragma unroll
  for (int j = 0; j < 48; ++j) {
    int kr = 4 * j + (khalf << 1);
    aF[j] = ld2(sP + am * 192 + kr);
  }

  auto loadX = [&](int r) {
    float* dst = sX + ((r + 3) % 3) * 580;
    bool zr = (r < 0) || (r >= L_) || (r >= len);
    const float* src =
        zr ? embs : (embs + ((size_t)bidx[p] * T_ + tidx[p * L_ + r]) * F_);
    for (int i = tid; i < 580; i += 256) {
      int w = i - 2;
      dst[i] = (zr || w < 0 || w >= F_) ? 0.f : src[w];
    }
  };

  auto prefX = [&](int r) {  // prefetch next gathered row into caches
    if (r < 0 || r >= L_ || r >= len) return;
    const float* src = embs + ((size_t)bidx[p] * T_ + tidx[p * L_ + r]) * F_;
    for (int i = tid; i < 36; i += 256) __builtin_prefetch(src + i * 16, 0, 1);
  };

  auto seluRow = [&](int h1) {
    float* S = sS + (h1 & 1) * (32 * 292);
    const float* x0 = sX + (((h1 - 1) + 3) % 3) * 580;
    const float* x1 = sX + ((h1 + 3) % 3) * 580;
    for (int i = tid; i < 32 * 290; i += 256) {
      int ci = i / 290, s = i % 290;
      int gc = wbase - 1 + s;  // conv1 output column
      float val = 0.f;
      if (gc >= 0 && gc < F_) {
        const float* wf = w1f + ci * 6;
        float acc = x0[gc + 1] * wf[0];
        acc = fmaf(x0[gc + 2], wf[1], acc);
        acc = fmaf(x0[gc + 3], wf[2], acc);
        acc = fmaf(x1[gc + 1], wf[3], acc);
        acc = fmaf(x1[gc + 2], wf[4], acc);
        acc = fmaf(x1[gc + 3], wf[5], acc);
        val = selu_f(fmaf(acc, bnA[ci], bnB[ci]));
      }
      S[(ci >> 1) * PW1 + 2 * s + (ci & 1)] = val;
    }
  };

  auto conv2Row = [&](int h) {
    const float* S0 = sS + (h & 1) * (32 * 292);
    const float* S1 = sS + ((h + 1) & 1) * (32 * 292);
    const float* xh = sX + ((h + 3) % 3) * 580;
    const int n = lane & 15;
    for (int nt = (wave >> 1); nt < 18; nt += 4) {
      const int w0 = nt * 16;
      v8f c = {0.f, 0.f, 0.f, 0.f, 0.f, 0.f, 0.f, 0.f};
#pragma unroll
      for (int j = 0; j < 48; ++j) {
        int kr = 4 * j + (khalf << 1);
        int ci = kr & 31, kh = (kr >> 5) & 1, kw = kr >> 6;
        v2f b = ld2((kh ? S1 : S0) + (ci >> 1) * PW1 + 2 * (w0 + n + kw));
        c = wmma4(aF[j], b, c);
      }
      const int w  = w0 + n;
      const int gw = wbase + w;
#pragma unroll
      for (int r = 0; r < 8; ++r) {
        int co = mtile * 16 + r + khalf * 8;
        float dsv = fmaf(xh[gw + 1], wdf[co * 3 + 0], bdf[co]);
        dsv = fmaf(xh[gw + 2], wdf[co * 3 + 1], dsv);
        dsv = fmaf(xh[gw + 3], wdf[co * 3 + 2], dsv);
        sP[co * 288 + w] = c[r] + b2f[co] + dsv;
      }
    }
  };

  loadX(h0 - 1);
  loadX(h0);
  __syncthreads();
  seluRow(h0);
  for (int h = h0; h < h0 + 32; ++h) {
    __syncthreads();
    loadX(h + 1);
    prefX(h + 2);
    __syncthreads();
    seluRow(h + 1);
    __syncthreads();
    conv2Row(h);
    __syncthreads();
    float mfh = (h < len) ? 1.f : 0.f;
    for (int i = tid; i < 32 * 96; i += 256) {
      int co = i / 96, wq = i % 96;
      const float* pr = sP + co * 288 + wq * 3;
      float mx = fmaxf(pr[0], fmaxf(pr[1], pr[2]));
      y1[(((size_t)p * 32 + co) * L_ + h) * L_ + wbase / 3 + wq] = mx * mfh;
    }
  }
}

// ============================ Block 2 ============================
// 32->64 ch, W=192; conv1/conv2/downsample all via WMMA; pool W/3;
// 7x7 adaptive-max; masked per-chunk partial sums of feat.
__global__ __launch_bounds__(256) void k_block2(
    const float* __restrict__ y1, const int* __restrict__ rlen,
    const float* __restrict__ w1, const float* __restrict__ b1,
    const float* __restrict__ bg, const float* __restrict__ bb,
    const float* __restrict__ bm, const float* __restrict__ bv,
    const float* __restrict__ w2, const float* __restrict__ b2,
    const float* __restrict__ wd, const float* __restrict__ bd,
    float* __restrict__ part) {
  extern __shared__ float sm2[];
  float* sA1  = sm2;                 // [64][192]
  float* sAd  = sA1 + B2_SA1;        // [64][96]
  float* sS   = sAd + B2_SAD;        // 2 slots, pair-interleaved
  float* sY   = sS + B2_SELU;        // 2 slots, pair-interleaved
  float* sP   = sY + B2_SY;          // [64][192]
  float* bnB2 = sP + B2_SPRE;
  float* b2f  = bnB2 + 64;
  float* bdf  = b2f + 64;
  float* fAcc = bdf + 64;            // [64] (49 used)

  const int tid = threadIdx.x, lane = tid & 31, wave = tid >> 5;
  const int p = blockIdx.x / 12, chunk = blockIdx.x % 12;
  const int h0 = chunk * 16;
  const int len = rlen[p];

  for (int i = tid; i < 64; i += 256) {
    float sc = bg[i] / sqrtf(bv[i] + 1e-5f);
    bnB2[i] = (b1[i] - bm[i]) * sc + bb[i];
    b2f[i] = b2[i];
    bdf[i] = bd[i];
  }
  if (tid < 64) fAcc[tid] = 0.f;
  // conv1 A staging (BN scale folded): k = kw*64 + kh*32 + ci
  for (int i = tid; i < 64 * 192; i += 256) {
    int co = i / 192, k = i % 192;
    int ci = k & 31, kh = (k >> 5) & 1, kw = k >> 6;
    float sc = bg[co] / sqrtf(bv[co] + 1e-5f);
    sA1[i] = w1[((co * 32 + ci) * 2 + kh) * 3 + kw] * sc;
  }
  // downsample A staging: k = kw*32 + ci
  for (int i = tid; i < 64 * 96; i += 256) {
    int co = i / 96, k = i % 96;
    int ci = k & 31, kw = k >> 5;
    sAd[i] = wd[(co * 32 + ci) * 3 + kw];
  }
  __syncthreads();

  const int mtile = wave & 3;
  const int am    = mtile * 16 + (lane & 15);
  const int khalf = lane >> 4;

  // conv2 A fragments in registers: k = kw*128 + kh*64 + ci
  v2f aF2[96];
#pragma unroll
  for (int j = 0; j < 96; ++j) {
    int kr = 4 * j + (khalf << 1);
    int ci = kr & 63, kh = (kr >> 6) & 1, kw = kr >> 7;
    const float* wp = w2 + ((am * 64 + ci) * 2 + kh) * 3 + kw;
    aF2[j].x = wp[0];
    aF2[j].y = wp[6];  // ci+1
  }

  auto loadY = [&](int r) {
    float* dst = sY + ((r + 2) & 1) * (32 * 196);
    bool zr = (r < 0) || (r >= L_);
    for (int i = tid; i < 32 * 196; i += 256) {
      int ci = i / 196, s = i % 196, col = s - 2;
      float v = 0.f;
      if (!zr && col >= 0 && col < L_)
        v = y1[(((size_t)p * 32 + ci) * L_ + r) * L_ + col];
      dst[(ci >> 1) * PW2 + 2 * s + (ci & 1)] = v;
    }
  };

  auto prefY = [&](int r) {  // prefetch next y1 row into caches
    if (r < 0 || r >= L_) return;
    for (int i = tid; i < 32 * 12; i += 256) {
      int ci = i / 12, seg = i % 12;
      __builtin_prefetch(y1 + (((size_t)p * 32 + ci) * L_ + r) * L_ + seg * 16,
                         0, 1);
    }
  };

  auto selu2Row = [&](int h1) {
    float* S = sS + (h1 & 1) * (64 * 196);
    const float* Y0 = sY + ((h1 + 1) & 1) * (32 * 196);  // y1 row h1-1
    const float* Y1 = sY + (h1 & 1) * (32 * 196);        // y1 row h1
    const int n = lane & 15;
    for (int nt = (wave >> 2); nt < 13; nt += 2) {
      int s0 = nt * 16;
      v8f c = {0.f, 0.f, 0.f, 0.f, 0.f, 0.f, 0.f, 0.f};
#pragma unroll
      for (int j = 0; j < 48; ++j) {
        int kr = 4 * j + (khalf << 1);
        int ci = kr & 31, kh = (kr >> 5) & 1, kw = kr >> 6;
        v2f a = ld2(sA1 + am * 192 + kr);
        v2f b = ld2((kh ? Y1 : Y0) + (ci >> 1) * PW2 + 2 * (s0 + n + kw));
        c = wmma4(a, b, c);
      }
      int s = s0 + n;
      if (s < 194) {
        int wc = s - 1;
        bool valid = (wc >= 0) && (wc < L_);
#pragma unroll
        for (int r = 0; r < 8; ++r) {
          int co = mtile * 16 + r + khalf * 8;
          S[(co >> 1) * PW2 + 2 * s + (co & 1)] =
              valid ? selu_f(c[r] + bnB2[co]) : 0.f;
        }
      }
    }
  };

  auto conv2Row = [&](int h) {
    const float* S0 = sS + (h & 1) * (64 * 196);
    const float* S1 = sS + ((h + 1) & 1) * (64 * 196);
    const float* Yh = sY + (h & 1) * (32 * 196);
    const int n = lane & 15;
    for (int nt = (wave >> 2); nt < 12; nt += 2) {
      int w0 = nt * 16;
      v8f c = {0.f, 0.f, 0.f, 0.f, 0.f, 0.f, 0.f, 0.f};
#pragma unroll
      for (int j = 0; j < 96; ++j) {  // conv2: K = 384
        int kr = 4 * j + (khalf << 1);
        int ci = kr & 63, kh = (kr >> 6) & 1, kw = kr >> 7;
        v2f b = ld2((kh ? S1 : S0) + (ci >> 1) * PW2 + 2 * (w0 + n + kw));
        c = wmma4(aF2[j], b, c);
      }
#pragma unroll
      for (int j = 0; j < 24; ++j) {  // downsample: K = 96
        int kr = 4 * j + (khalf << 1);
        int ci = kr & 31, kw = kr >> 5;
        v2f a = ld2(sAd + am * 96 + kr);
        v2f b = ld2(Yh + (ci >> 1) * PW2 + 2 * (w0 + n + kw + 1));
        c = wmma4(a, b, c);
      }
      int w = w0 + n;
#pragma unroll
      for (int r = 0; r < 8; ++r) {
        int co = mtile * 16 + r + khalf * 8;
        sP[co * 192 + w] = c[r] + b2f[co] + bdf[co];
      }
    }
  };

  loadY(h0 - 1);
  loadY(h0);
  __syncthreads();
  selu2Row(h0);
  for (int h = h0; h < h0 + 16; ++h) {
    __syncthreads();
    loadY(h + 1);
    prefY(h + 2);
    __syncthreads();
    selu2Row(h + 1);
    __syncthreads();
    conv2Row(h);
    __syncthreads();
    if (h < len && tid < 49) {
      int bi = tid / 7, bj = tid % 7;
      int c0 = (bi * 64) / 7, c1 = ((bi + 1) * 64 + 6) / 7;
      int g0 = (bj * 64) / 7, g1 = ((bj + 1) * 64 + 6) / 7;
      float mx = -3.402823466e38f;
      for (int c = c0; c < c1; ++c)
        for (int w = g0 * 3; w < g1 * 3; ++w)  // max over pre-pool = pooled bin max
          mx = fmaxf(mx, sP[c * 192 + w]);
      fAcc[tid] += mx;
    }
  }
  __syncthreads();
  if (tid < 49) part[((size_t)p * 12 + chunk) * 49 + tid] = fAcc[tid];
}

// ============================ Head ============================
__global__ __launch_bounds__(256) void k_head(
    const float* __restrict__ part, const float* __restrict__ w6,
    const float* __restrict__ b6, const float* __restrict__ w7,
    const float* __restrict__ b7, const float* __restrict__ wc,
    const float* __restrict__ bc, const float* __restrict__ wb,
    const float* __restrict__ bbx, float* __restrict__ out) {
  __shared__ float sf[49];
  __shared__ float sh[256];
  const int tid = threadIdx.x;
  const int p = blockIdx.x;
  if (tid < 49) {
    float s = 0.f;
    for (int c = 0; c < 12; ++c) s += part[((size_t)p * 12 + c) * 49 + tid];
    sf[tid] = s;
  }
  __syncthreads();
  float a = b6[tid];
  for (int k = 0; k < 49; ++k) a = fmaf(w6[tid * 49 + k], sf[k], a);
  sh[tid] = fmaxf(a, 0.f);
  __syncthreads();
  float h7 = b7[tid];
  for (int k = 0; k < 256; ++k) h7 = fmaf(w7[tid * 256 + k], sh[k], h7);
  h7 = fmaxf(h7, 0.f);
  __syncthreads();
  sh[tid] = h7;
  __syncthreads();
  if (tid == 0) {
    float s = bc[0];
    for (int k = 0; k < 256; ++k) s = fmaf(wc[k], sh[k], s);
    out[p] = 1.f / (1.f + __expf(-s));
  } else if (tid < 3) {
    int r = tid - 1;
    float s = bbx[r];
    for (int k = 0; k < 256; ++k) s = fmaf(wb[r * 256 + k], sh[k], s);
    out[32 + p * 2 + r] = s;
  }
}

extern "C" void kernel_launch(void* const* d_in, const int* in_sizes, int n_in,
                              void* d_out, int out_size, void* d_ws,
                              size_t ws_size, hipStream_t stream) {
  (void)in_sizes; (void)n_in; (void)out_size; (void)ws_size;
  const float* embs = (const float*)d_in[0];
  const int* bidx   = (const int*)d_in[1];
  const int* tidx   = (const int*)d_in[2];
  const int* rlen   = (const int*)d_in[3];
  const float* c1_w1 = (const float*)d_in[4];
  const float* c1_b1 = (const float*)d_in[5];
  const float* c1_g  = (const float*)d_in[6];
  const float* c1_be = (const float*)d_in[7];
  const float* c1_m  = (const float*)d_in[8];
  const float* c1_v  = (const float*)d_in[9];
  const float* c1_w2 = (const float*)d_in[10];
  const float* c1_b2 = (const float*)d_in[11];
  const float* c1_wd = (const float*)d_in[12];
  const float* c1_bd = (const float*)d_in[13];
  const float* c2_w1 = (const float*)d_in[14];
  const float* c2_b1 = (const float*)d_in[15];
  const float* c2_g  = (const float*)d_in[16];
  const float* c2_be = (const float*)d_in[17];
  const float* c2_m  = (const float*)d_in[18];
  const float* c2_v  = (const float*)d_in[19];
  const float* c2_w2 = (const float*)d_in[20];
  const float* c2_b2 = (const float*)d_in[21];
  const float* c2_wd = (const float*)d_in[22];
  const float* c2_bd = (const float*)d_in[23];
  const float* fc6_w = (const float*)d_in[24];
  const float* fc6_b = (const float*)d_in[25];
  const float* fc7_w = (const float*)d_in[26];
  const float* fc7_b = (const float*)d_in[27];
  const float* cls_w = (const float*)d_in[28];
  const float* cls_b = (const float*)d_in[29];
  const float* box_w = (const float*)d_in[30];
  const float* box_b = (const float*)d_in[31];

  float* y1   = (float*)d_ws;
  float* part = y1 + Y1_FLOATS;
  float* out  = (float*)d_out;

  const int lds1 = (int)(B1_LDS * sizeof(float));
  const int lds2 = (int)(B2_LDS * sizeof(float));
  (void)hipFuncSetAttribute((const void*)k_block1,
                            hipFuncAttributeMaxDynamicSharedMemorySize, lds1);
  (void)hipFuncSetAttribute((const void*)k_block2,
                            hipFuncAttributeMaxDynamicSharedMemorySize, lds2);

  k_block1<<<dim3(32 * 6 * 2), dim3(256), lds1, stream>>>(
      embs, bidx, tidx, rlen, c1_w1, c1_b1, c1_g, c1_be, c1_m, c1_v, c1_w2,
      c1_b2, c1_wd, c1_bd, y1);
  k_block2<<<dim3(32 * 12), dim3(256), lds2, stream>>>(
      y1, rlen, c2_w1, c2_b1, c2_g, c2_be, c2_m, c2_v, c2_w2, c2_b2, c2_wd,
      c2_bd, part);
  k_head<<<dim3(32), dim3(256), 0, stream>>>(part, fc6_w, fc6_b, fc7_w, fc7_b,
                                             cls_w, cls_b, box_w, box_b, out);
}